// DecoderWithKVCache_37890201485354
// MI455X (gfx1250) — compile-verified
//
#include <hip/hip_runtime.h>

#define LN_LAYERS 12
#define NH 12
#define HD 64
#define DM 768
#define FF 3072
#define NV 51865
#define NENC 1500
#define SQ 256
#define ATT_SCALE 0.125f

typedef __attribute__((ext_vector_type(2))) float v2f;
typedef __attribute__((ext_vector_type(8))) float v8f;

// ---------------------------------------------------------------- utilities
__device__ __forceinline__ float wave_reduce_sum(float v) {
  #pragma unroll
  for (int off = 16; off > 0; off >>= 1) v += __shfl_xor(v, off, 32);
  return v;
}

// ---------------------------------------------------------------- embed
__global__ void embed_kernel(const int* __restrict__ ids,
                             const float* __restrict__ embed_w,
                             float* __restrict__ h) {
  int t = blockIdx.x * blockDim.x + threadIdx.x;
  if (t < DM) h[t] = embed_w[(size_t)ids[0] * DM + t];
}

// ---------------------------------------------------------------- layernorm (1 block / 256 thr)
__global__ void layernorm_kernel(const float* __restrict__ x,
                                 const float* __restrict__ g,
                                 const float* __restrict__ b,
                                 float* __restrict__ y) {
  __shared__ float s1[256];
  __shared__ float s2[256];
  int t = threadIdx.x;
  float sum = 0.f, sq = 0.f;
  for (int i = t; i < DM; i += 256) { float v = x[i]; sum += v; sq += v * v; }
  s1[t] = sum; s2[t] = sq;
  __syncthreads();
  for (int o = 128; o > 0; o >>= 1) {
    if (t < o) { s1[t] += s1[t + o]; s2[t] += s2[t + o]; }
    __syncthreads();
  }
  float mu  = s1[0] * (1.f / DM);
  float var = s2[0] * (1.f / DM) - mu * mu;
  float r   = rsqrtf(var + 1e-5f);
  for (int i = t; i < DM; i += 256) y[i] = (x[i] - mu) * r * g[i] + b[i];
}

// ---------------------------------------------------------------- GEMV: y = [res +] W x + bias, optional relu
// W row-major (N,K), K % 128 == 0. One wave per output row; float4 (b128)
// loads -> 512 B per wave-iteration, HBM-roofline streaming.
__global__ void gemv_kernel(const float* __restrict__ W,
                            const float* __restrict__ bias,
                            const float* __restrict__ x,
                            const float* __restrict__ res,
                            float* __restrict__ y,
                            int N, int K, int relu) {
  int lane = threadIdx.x & 31;
  int n = blockIdx.x * (blockDim.x >> 5) + (threadIdx.x >> 5);
  if (n >= N) return;
  const float4* Wr = (const float4*)(W + (size_t)n * K);
  const float4* xv = (const float4*)x;
  int kq = K >> 2;  // float4 count
  float acc = 0.f;
  for (int i = lane; i < kq; i += 32) {
    float4 w = Wr[i];
    float4 v = xv[i];
    acc = fmaf(w.x, v.x, acc);
    acc = fmaf(w.y, v.y, acc);
    acc = fmaf(w.z, v.z, acc);
    acc = fmaf(w.w, v.w, acc);
  }
  acc = wave_reduce_sum(acc);
  if (lane == 0) {
    float v = acc;
    if (bias) v += bias[n];
    if (res)  v += res[n];
    if (relu) v = fmaxf(v, 0.f);
    y[n] = v;
  }
}

// ---------------------------------------------------------------- write new k,v into cache at cp
__global__ void kv_update_kernel(float* __restrict__ new_ks, float* __restrict__ new_vs,
                                 const float* __restrict__ k, const float* __restrict__ v,
                                 const int* __restrict__ cp_ptr, int layer) {
  int t = blockIdx.x * blockDim.x + threadIdx.x;
  if (t >= DM) return;
  int cp = cp_ptr[0];
  int h = t >> 6, hd = t & 63;
  size_t idx = ((((size_t)layer * NH + h) * SQ) + cp) * HD + hd;
  new_ks[idx] = k[t];
  new_vs[idx] = v[t];
}

// ---------------------------------------------------------------- self attention (grid = NH, 256 thr)
__global__ void self_attn_kernel(const float* __restrict__ q,
                                 const float* __restrict__ Kc,
                                 const float* __restrict__ Vc,
                                 float* __restrict__ ao) {
  __shared__ float aw[SQ];
  __shared__ float red[256];
  int h = blockIdx.x, t = threadIdx.x;
  const float4* qh = (const float4*)(q + h * HD);
  const float4* kr = (const float4*)(Kc + ((size_t)h * SQ + t) * HD);
  float s = 0.f;
  #pragma unroll
  for (int d = 0; d < HD / 4; ++d) {
    float4 a = qh[d], b = kr[d];
    s = fmaf(a.x, b.x, s); s = fmaf(a.y, b.y, s);
    s = fmaf(a.z, b.z, s); s = fmaf(a.w, b.w, s);
  }
  s *= ATT_SCALE;
  red[t] = s; __syncthreads();
  for (int o = 128; o > 0; o >>= 1) { if (t < o) red[t] = fmaxf(red[t], red[t + o]); __syncthreads(); }
  float m = red[0]; __syncthreads();
  float e = __expf(s - m);
  aw[t] = e; red[t] = e; __syncthreads();
  for (int o = 128; o > 0; o >>= 1) { if (t < o) red[t] += red[t + o]; __syncthreads(); }
  float inv = 1.f / red[0]; __syncthreads();
  int hd = t & 63, grp = t >> 6;
  const float* Vh = Vc + (size_t)h * SQ * HD;
  float p = 0.f;
  for (int sidx = grp * 64; sidx < grp * 64 + 64; ++sidx) p += aw[sidx] * Vh[(size_t)sidx * HD + hd];
  red[t] = p; __syncthreads();
  if (t < 64) ao[h * HD + t] = inv * (red[t] + red[t + 64] + red[t + 128] + red[t + 192]);
}

// ---------------------------------------------------------------- cross attention (grid = NH, 256 thr)
// ek/ev: row-major (NENC, DM); head h lives at column h*HD
__global__ void cross_attn_kernel(const float* __restrict__ q,
                                  const float* __restrict__ ek,
                                  const float* __restrict__ ev,
                                  float* __restrict__ ao) {
  __shared__ float aw[1536];
  __shared__ float red[256];
  int h = blockIdx.x, t = threadIdx.x;
  const float4* qh = (const float4*)(q + h * HD);
  for (int e = t; e < 1536; e += 256) {
    float s = -1e30f;
    if (e < NENC) {
      const float4* kr = (const float4*)(ek + (size_t)e * DM + h * HD);
      float acc = 0.f;
      #pragma unroll
      for (int d = 0; d < HD / 4; ++d) {
        float4 a = qh[d], b = kr[d];
        acc = fmaf(a.x, b.x, acc); acc = fmaf(a.y, b.y, acc);
        acc = fmaf(a.z, b.z, acc); acc = fmaf(a.w, b.w, acc);
      }
      s = acc * ATT_SCALE;
    }
    aw[e] = s;
  }
  __syncthreads();
  float m = -1e30f;
  for (int e = t; e < 1536; e += 256) m = fmaxf(m, aw[e]);
  red[t] = m; __syncthreads();
  for (int o = 128; o > 0; o >>= 1) { if (t < o) red[t] = fmaxf(red[t], red[t + o]); __syncthreads(); }
  m = red[0]; __syncthreads();
  float sum = 0.f;
  for (int e = t; e < 1536; e += 256) {
    float ex = (e < NENC) ? __expf(aw[e] - m) : 0.f;
    aw[e] = ex; sum += ex;
  }
  red[t] = sum; __syncthreads();
  for (int o = 128; o > 0; o >>= 1) { if (t < o) red[t] += red[t + o]; __syncthreads(); }
  float inv = 1.f / red[0]; __syncthreads();
  int hd = t & 63, grp = t >> 6;
  float p = 0.f;
  for (int e = grp; e < NENC; e += 4) p += aw[e] * ev[(size_t)e * DM + h * HD + hd];
  red[t] = p; __syncthreads();
  if (t < 64) ao[h * HD + t] = inv * (red[t] + red[t + 64] + red[t + 128] + red[t + 192]);
}

// ---------------------------------------------------------------- encoder K/V projection GEMM via WMMA fp32
// out(e,n) = sum_k enc(e,k) * W(n,k) + bias(n)   [out = enc @ W.T + b]
// grid: (ceil(NENC/32), DM/64, 2*nLayers); block: 32 (one wave).
// blockIdx.z: (z>>1) = layer slice of W/bias, (z&1) = K vs V matrix;
// output tile -> outbase + z*NENC*DM.
// Each wave computes a 32x64 tile: two 16-row A fragments share four B
// fragments -> 8 v_wmma_f32_16x16x4_f32 per 6 b64 loads per k-step.
__global__ void enc_kv_wmma_kernel(const float* __restrict__ enc,
                                   const float* __restrict__ Wk_base, const float* __restrict__ bk_base,
                                   const float* __restrict__ Wv_base, const float* __restrict__ bv_base,
                                   float* __restrict__ outbase) {
  int z = blockIdx.z;
  int lsel = z >> 1;
  int kv   = z & 1;
  const float* W    = (kv ? Wv_base : Wk_base) + (size_t)lsel * DM * DM;
  const float* bias = (kv ? bv_base : bk_base) + (size_t)lsel * DM;
  float*       out  = outbase + (size_t)z * NENC * DM;

  int lane = threadIdx.x;
  int m0 = blockIdx.x * 32;
  int n0 = blockIdx.y * 64;
  int row = lane & 15;              // A: M = lane%16
  int kl  = (lane >> 4) * 2;        // A/B: K sub-offset per lane half
  int col = lane & 15;              // B/D: N = lane%16

  v8f acc[2][4] = {};
  int mA = m0 + row;
  int mB = m0 + 16 + row;
  bool vA = (mA < NENC);
  bool vB = (mB < NENC);
  const float* arowA = enc + (size_t)mA * DM;
  const float* arowB = enc + (size_t)mB * DM;

  const float* wrow0 = W + (size_t)(n0 + col) * DM;
  const float* wrow1 = wrow0 + (size_t)16 * DM;
  const float* wrow2 = wrow0 + (size_t)32 * DM;
  const float* wrow3 = wrow0 + (size_t)48 * DM;

  for (int k0 = 0; k0 < DM; k0 += 4) {
    int kk = k0 + kl;
    v2f a0 = vA ? *(const v2f*)(arowA + kk) : (v2f){0.f, 0.f};
    v2f a1 = vB ? *(const v2f*)(arowB + kk) : (v2f){0.f, 0.f};
    v2f b0 = *(const v2f*)(wrow0 + kk);
    v2f b1 = *(const v2f*)(wrow1 + kk);
    v2f b2 = *(const v2f*)(wrow2 + kk);
    v2f b3 = *(const v2f*)(wrow3 + kk);
    acc[0][0] = __builtin_amdgcn_wmma_f32_16x16x4_f32(false, a0, false, b0, (short)0, acc[0][0], false, false);
    acc[1][0] = __builtin_amdgcn_wmma_f32_16x16x4_f32(false, a1, false, b0, (short)0, acc[1][0], false, false);
    acc[0][1] = __builtin_amdgcn_wmma_f32_16x16x4_f32(false, a0, false, b1, (short)0, acc[0][1], false, false);
    acc[1][1] = __builtin_amdgcn_wmma_f32_16x16x4_f32(false, a1, false, b1, (short)0, acc[1][1], false, false);
    acc[0][2] = __builtin_amdgcn_wmma_f32_16x16x4_f32(false, a0, false, b2, (short)0, acc[0][2], false, false);
    acc[1][2] = __builtin_amdgcn_wmma_f32_16x16x4_f32(false, a1, false, b2, (short)0, acc[1][2], false, false);
    acc[0][3] = __builtin_amdgcn_wmma_f32_16x16x4_f32(false, a0, false, b3, (short)0, acc[0][3], false, false);
    acc[1][3] = __builtin_amdgcn_wmma_f32_16x16x4_f32(false, a1, false, b3, (short)0, acc[1][3], false, false);
  }

  // D layout: lane = col (lane%16), VGPR v -> row v + 8*(lane>=16)
  #pragma unroll
  for (int tl = 0; tl < 2; ++tl) {
    int mbase = m0 + tl * 16 + ((lane >> 4) * 8);
    #pragma unroll
    for (int j = 0; j < 4; ++j) {
      int n = n0 + j * 16 + col;
      float bn = bias[n];
      #pragma unroll
      for (int v = 0; v < 8; ++v) {
        int mm = mbase + v;
        if (mm < NENC) out[(size_t)mm * DM + n] = acc[tl][j][v] + bn;
      }
    }
  }
}

// ---------------------------------------------------------------- launch
extern "C" void kernel_launch(void* const* d_in, const int* in_sizes, int n_in,
                              void* d_out, int out_size, void* d_ws, size_t ws_size,
                              hipStream_t stream) {
  const int*   input_ids = (const int*)d_in[0];
  const float* enc       = (const float*)d_in[1];
  const float* past_ks   = (const float*)d_in[3];
  const float* past_vs   = (const float*)d_in[4];
  const int*   cache_pos = (const int*)d_in[5];
  const float* embed_w   = (const float*)d_in[6];
  const float* lm_head_w = (const float*)d_in[7];
  const float* lnf_g     = (const float*)d_in[8];
  const float* lnf_b     = (const float*)d_in[9];
  const float* ln1_g     = (const float*)d_in[10];
  const float* ln1_b     = (const float*)d_in[11];
  const float* wq = (const float*)d_in[12]; const float* bq = (const float*)d_in[13];
  const float* wk = (const float*)d_in[14]; const float* bk = (const float*)d_in[15];
  const float* wv = (const float*)d_in[16]; const float* bv = (const float*)d_in[17];
  const float* wo = (const float*)d_in[18]; const float* bo = (const float*)d_in[19];
  const float* ln2_g = (const float*)d_in[20]; const float* ln2_b = (const float*)d_in[21];
  const float* cwq = (const float*)d_in[22]; const float* cbq = (const float*)d_in[23];
  const float* cwk = (const float*)d_in[24]; const float* cbk = (const float*)d_in[25];
  const float* cwv = (const float*)d_in[26]; const float* cbv = (const float*)d_in[27];
  const float* cwo = (const float*)d_in[28]; const float* cbo = (const float*)d_in[29];
  const float* ln3_g = (const float*)d_in[30]; const float* ln3_b = (const float*)d_in[31];
  const float* w1 = (const float*)d_in[32]; const float* b1 = (const float*)d_in[33];
  const float* w2 = (const float*)d_in[34]; const float* b2 = (const float*)d_in[35];

  const size_t kv_elems = (size_t)LN_LAYERS * NH * SQ * HD;
  float* logits = (float*)d_out;
  float* new_ks = logits + NV;
  float* new_vs = new_ks + kv_elems;

  float* ws   = (float*)d_ws;
  float* h    = ws;
  float* hn   = ws + DM;
  float* q    = ws + 2 * DM;
  float* ao   = ws + 3 * DM;
  float* kbuf = ws + 4 * DM;
  float* vbuf = ws + 5 * DM;
  float* ffn  = ws + 6 * DM;              // FF floats
  float* ekv  = ws + 6 * DM + FF;         // enc K/V buffers start here

  // Batched path keeps all 24 (layer x {K,V}) encoder projections resident:
  // 24 * NENC * DM floats (~110.6 MB). Fallback: 2 buffers (~9.2 MB),
  // recomputed per layer. Path depends only on ws_size -> deterministic.
  const size_t small_elems   = (size_t)6 * DM + FF;
  const size_t batched_bytes = (small_elems + (size_t)2 * LN_LAYERS * NENC * DM) * sizeof(float);
  const bool batched = (ws_size >= batched_bytes);

  hipMemcpyAsync(new_ks, past_ks, kv_elems * sizeof(float), hipMemcpyDeviceToDevice, stream);
  hipMemcpyAsync(new_vs, past_vs, kv_elems * sizeof(float), hipMemcpyDeviceToDevice, stream);

  // All-layer encoder K/V projection up front (independent of decoder state):
  // one launch, 47*12*24 = 13536 waves -> full latency hiding.
  if (batched) {
    dim3 encGridAll((NENC + 31) / 32, DM / 64, 2 * LN_LAYERS);
    enc_kv_wmma_kernel<<<encGridAll, 32, 0, stream>>>(enc, cwk, cbk, cwv, cbv, ekv);
  }

  embed_kernel<<<(DM + 255) / 256, 256, 0, stream>>>(input_ids, embed_w, h);

  const int gD = (DM + 7) / 8;
  const int gF = (FF + 7) / 8;

  for (int l = 0; l < LN_LAYERS; ++l) {
    size_t mo = (size_t)l * DM * DM;   // D x D matrices
    size_t vo = (size_t)l * DM;        // D vectors
    size_t fo = (size_t)l * FF;        // F vectors

    // ---- self attention ----
    layernorm_kernel<<<1, 256, 0, stream>>>(h, ln1_g + vo, ln1_b + vo, hn);
    gemv_kernel<<<gD, 256, 0, stream>>>(wq + mo, bq + vo, hn, nullptr, q,    DM, DM, 0);
    gemv_kernel<<<gD, 256, 0, stream>>>(wk + mo, bk + vo, hn, nullptr, kbuf, DM, DM, 0);
    gemv_kernel<<<gD, 256, 0, stream>>>(wv + mo, bv + vo, hn, nullptr, vbuf, DM, DM, 0);
    kv_update_kernel<<<(DM + 255) / 256, 256, 0, stream>>>(new_ks, new_vs, kbuf, vbuf, cache_pos, l);
    self_attn_kernel<<<NH, 256, 0, stream>>>(q, new_ks + (size_t)l * NH * SQ * HD,
                                             new_vs + (size_t)l * NH * SQ * HD, ao);
    gemv_kernel<<<gD, 256, 0, stream>>>(wo + mo, bo + vo, ao, h, h, DM, DM, 0);

    // ---- cross attention ----
    layernorm_kernel<<<1, 256, 0, stream>>>(h, ln2_g + vo, ln2_b + vo, hn);
    gemv_kernel<<<gD, 256, 0, stream>>>(cwq + mo, cbq + vo, hn, nullptr, q, DM, DM, 0);
    const float* ekb;
    const float* evb;
    if (batched) {
      ekb = ekv + (size_t)(2 * l)     * NENC * DM;
      evb = ekv + (size_t)(2 * l + 1) * NENC * DM;
    } else {
      dim3 encGrid((NENC + 31) / 32, DM / 64, 2);
      enc_kv_wmma_kernel<<<encGrid, 32, 0, stream>>>(enc, cwk + mo, cbk + vo,
                                                     cwv + mo, cbv + vo, ekv);
      ekb = ekv;
      evb = ekv + (size_t)NENC * DM;
    }
    cross_attn_kernel<<<NH, 256, 0, stream>>>(q, ekb, evb, ao);
    gemv_kernel<<<gD, 256, 0, stream>>>(cwo + mo, cbo + vo, ao, h, h, DM, DM, 0);

    // ---- FFN ----
    layernorm_kernel<<<1, 256, 0, stream>>>(h, ln3_g + vo, ln3_b + vo, hn);
    gemv_kernel<<<gF, 256, 0, stream>>>(w1 + (size_t)l * FF * DM, b1 + fo, hn, nullptr, ffn, FF, DM, 1);
    gemv_kernel<<<gD, 256, 0, stream>>>(w2 + (size_t)l * DM * FF, b2 + vo, ffn, h, h, DM, FF, 0);
  }

  layernorm_kernel<<<1, 256, 0, stream>>>(h, lnf_g, lnf_b, hn);
  gemv_kernel<<<(NV + 7) / 8, 256, 0, stream>>>(lm_head_w, nullptr, hn, nullptr, logits, NV, DM, 0);
}